// ScaleLayer_54906861912422
// MI455X (gfx1250) — compile-verified
//
#include <hip/hip_runtime.h>
#include <math.h>

// y[b,d] = x[b,d] * exp(diag[d]); second output = diag passthrough.
// Pure streaming kernel: 512 MB of traffic, ~22 us floor at 23.3 TB/s.

typedef float v4f __attribute__((ext_vector_type(4)));

#define THREADS 256
#define DCOLS 4096
#define SLOTS (DCOLS / 4)                      // 1024 float4 scale slots (16 KB LDS)
#define SLOTS_PER_THREAD (SLOTS / THREADS)     // 4
#define ROWS_PER_BLOCK 8
#define F4_PER_BLOCK (ROWS_PER_BLOCK * DCOLS / 4)  // 8192 float4 per block
#define ITERS (F4_PER_BLOCK / THREADS)             // 32

// LDS byte offset of a __shared__ object (generic -> addrspace(3) -> int).
__device__ static inline unsigned lds_byte_offset(const void* p) {
  return (unsigned)(size_t)(__attribute__((address_space(3))) const void*)p;
}

__global__ __launch_bounds__(THREADS)
void ScaleLayer_scale_rows(const float* __restrict__ x,
                           const float* __restrict__ diag,
                           float* __restrict__ out)
{
  __shared__ v4f s_scale[SLOTS];
  const int t = threadIdx.x;

  // ---- Stage raw diag into LDS with CDNA5 async global->LDS copies ----
  // 1024 float4 slots, 256 threads x 4 slots each; tracked by ASYNCcnt.
  const unsigned lds_base = lds_byte_offset(&s_scale[0]);
#pragma unroll
  for (int k = 0; k < SLOTS_PER_THREAD; ++k) {
    const int slot = t + k * THREADS;
    const unsigned loff = lds_base + (unsigned)slot * 16u;
    const float* gaddr = diag + slot * 4;
    asm volatile("global_load_async_to_lds_b128 %0, %1, off"
                 :: "v"(loff), "v"(gaddr)
                 : "memory");
  }
  // Each wave drains its own async copies, then barrier makes all visible.
  asm volatile("s_wait_asynccnt 0" ::: "memory");
  __syncthreads();

  // ---- exp() in place: each thread owns its slots, no cross-thread hazard ----
#pragma unroll
  for (int k = 0; k < SLOTS_PER_THREAD; ++k) {
    const int slot = t + k * THREADS;
    v4f v = s_scale[slot];
    v4f e;
    e.x = expf(v.x); e.y = expf(v.y); e.z = expf(v.z); e.w = expf(v.w);
    s_scale[slot] = e;
  }
  __syncthreads();

  // ---- Stream 8 rows per block with NT b128 loads/stores ----
  const size_t base_f4 = (size_t)blockIdx.x * F4_PER_BLOCK;
  const v4f* __restrict__ xv = (const v4f*)x + base_f4;
  v4f* __restrict__ ov = (v4f*)out + base_f4;

#pragma unroll
  for (int i = 0; i < ITERS; ++i) {
    const int j = i * THREADS + t;       // 0..8191, fully coalesced
    const int c4 = j & (SLOTS - 1);      // column float4 index (D is pow2-multiple of 4)
    v4f xd = __builtin_nontemporal_load(xv + j);   // TH=NT: single-touch stream
    v4f r = xd * s_scale[c4];                      // ds_load_b128 broadcast scale
    __builtin_nontemporal_store(r, ov + j);        // TH=NT: never re-read
  }
}

__global__ __launch_bounds__(THREADS)
void ScaleLayer_copy_diag(const float* __restrict__ diag,
                          float* __restrict__ out_tail)
{
  const int i = blockIdx.x * blockDim.x + threadIdx.x;
  if (i < DCOLS) out_tail[i] = diag[i];
}

extern "C" void kernel_launch(void* const* d_in, const int* in_sizes, int n_in,
                              void* d_out, int out_size, void* d_ws, size_t ws_size,
                              hipStream_t stream) {
  const float* x    = (const float*)d_in[0];   // [B, 4096] fp32
  const float* diag = (const float*)d_in[1];   // [4096] fp32
  float* out = (float*)d_out;                  // [B*4096] out ++ [4096] diag

  const int n_x  = in_sizes[0];                // B * 4096
  const int rows = n_x / DCOLS;                // B = 16384
  const int blocks = rows / ROWS_PER_BLOCK;    // 2048

  ScaleLayer_scale_rows<<<blocks, THREADS, 0, stream>>>(x, diag, out);
  ScaleLayer_copy_diag<<<(DCOLS + THREADS - 1) / THREADS, THREADS, 0, stream>>>(
      diag, out + (size_t)n_x);
}